// ConvCNP_53343493817017
// MI455X (gfx1250) — compile-verified
//
#include <hip/hip_runtime.h>
#include <hip/hip_bf16.h>
#include <math.h>
#include <stdint.h>

typedef __attribute__((ext_vector_type(2))) float v2f;
typedef __attribute__((ext_vector_type(8))) float v8f;

#define G_GRID 1000
#define BATCH  32
#define NC     2048
#define NT     2048
#define STR    1024           // padded row stride (floats) for activation rows
#define PADL   8              // left zero-pad inside LDS rows (conv halo)
#define LOG2E  1.4426950408889634f
#define LOG_2PI 1.8378770664093453f

// LDS layout for conv kernel (float offsets)
#define OFF_A16 0             // 16 ch x STR
#define OFF_A32 16384         // 32 ch x STR
#define OFF_H2  49152         // 2  ch x STR (layer-1 input)
#define OFF_WB  51200         // weight im2col buffer (max 2560 floats)
#define OFF_BI  53760         // 32 bias floats
#define SMEM_FLOATS 53792

// ---------------------------------------------------------------------------
// CDNA5 async global->LDS copy (ASYNCcnt-tracked, no VGPR round trip).
// LDS operand = low 32 bits of the generic pointer (aperture mapping: the
// flat LDS address's [31:0] is the wave-logical LDS byte address).
// ---------------------------------------------------------------------------
__device__ __forceinline__ void async_ld_b32(void* lds_dst, const float* gsrc) {
    uint32_t la = (uint32_t)(uintptr_t)lds_dst;
    asm volatile("global_load_async_to_lds_b32 %0, %1, off"
                 :: "v"(la), "v"(gsrc) : "memory");
}
__device__ __forceinline__ void wait_async0() {
#if defined(__has_builtin)
#if __has_builtin(__builtin_amdgcn_s_wait_asynccnt)
    __builtin_amdgcn_s_wait_asynccnt(0);
    return;
#else
    asm volatile("s_wait_asynccnt 0x0" ::: "memory");
    return;
#endif
#else
    asm volatile("s_wait_asynccnt 0x0" ::: "memory");
#endif
}

// ---------------------------------------------------------------------------
// Kernel 1: encode set-conv.  h0[g] = sum_n w, h1[g] = (sum_n w*y)/(h0+1e-8)
// w = exp(-(t_g - x_n)^2 / (2 ls^2)) = exp2((t-x)^2 * c2), double-buffered
// async staging of (x,y) context chunks overlapped with the exp/FMA loop.
// Grid: (8, B), block 128 -> one g per thread (valid g<1000).
// ---------------------------------------------------------------------------
__global__ void encode_kernel(const float* __restrict__ xc,
                              const float* __restrict__ yc,
                              const float* __restrict__ ls_x_p,
                              float* __restrict__ hbuf) {
    __shared__ float xs[2][128];
    __shared__ float ys[2][128];
    const int b   = blockIdx.y;
    const int tid = threadIdx.x;
    const int g   = blockIdx.x * 128 + tid;
    const float ls = ls_x_p[0];
    const float c2 = -LOG2E / (2.0f * ls * ls);
    const float tg = -2.2f + 4.4f * (float)g / (float)(G_GRID - 1);
    const float* xb = xc + b * NC;
    const float* yb = yc + b * NC;
    float s0 = 0.0f, s1 = 0.0f;

    const int nchunks = NC / 128;
    async_ld_b32(&xs[0][tid], xb + tid);          // prime the pipeline
    async_ld_b32(&ys[0][tid], yb + tid);
    for (int it = 0; it < nchunks; ++it) {
        const int cur = it & 1;
        wait_async0();                            // chunk `it` landed (this wave)
        __syncthreads();                          // all waves: data in, buffers free
        if (it + 1 < nchunks) {                   // stream chunk it+1 behind compute
            async_ld_b32(&xs[cur ^ 1][tid], xb + (it + 1) * 128 + tid);
            async_ld_b32(&ys[cur ^ 1][tid], yb + (it + 1) * 128 + tid);
        }
#pragma unroll 4
        for (int j = 0; j < 128; ++j) {
            float dx = tg - xs[cur][j];
            float w  = exp2f(dx * dx * c2);
            s0 += w;
            s1 += w * ys[cur][j];
        }
    }
    if (g < G_GRID) {
        hbuf[(b * 2 + 0) * STR + g] = s0;
        hbuf[(b * 2 + 1) * STR + g] = s1 / (s0 + 1e-8f);
    }
}

// ---------------------------------------------------------------------------
// Conv layer as GEMM on V_WMMA_F32_16X16X4_F32 (full fp32 path).
// out[co,g] = relu( sum_{kk=ci*5+k} W[co,kk] * in[ci, g+k-2] + bias[co] )
// A tile: 16 Cout-rows x 4 K (LDS, K zero-padded to x4)
// B tile: 4 K x 16 g         (gathered from LDS activation rows, k-2 shift)
// Lane map per ISA: A/B VGPR0 <-> K=2*half, VGPR1 <-> K=2*half+1;
// A row = lane&15, B/C/D col = lane&15, C/D row r -> r+8*half.
// ---------------------------------------------------------------------------
__device__ __forceinline__ void conv_layer(
    float* smem, int in_off, int Cin, int out_off, int Cout, int CoutP,
    const float* __restrict__ Wg, const float* __restrict__ bg,
    bool relu, bool last, float* __restrict__ yg_out, int b) {

    const int tid = threadIdx.x;
    const int K   = Cin * 5;
    const int Kp  = (K + 3) & ~3;
    const int nkc = Kp >> 2;
    float* WB = smem + OFF_WB;
    float* BI = smem + OFF_BI;

    __syncthreads();                       // previous layer fully consumed
    for (int i = tid; i < CoutP * Kp; i += 256) WB[i] = 0.0f;
    if (tid < 32) BI[tid] = (tid < Cout) ? bg[tid] : 0.0f;
    __syncthreads();
    for (int i = tid; i < Cout * K; i += 256) {
        int co = i / K, kk = i - co * K;   // W is (Cout, Cin, 5) row-major
        WB[co * Kp + kk] = Wg[i];
    }
    __syncthreads();

    const int lane = tid & 31;
    const int half = lane >> 4;
    const int col  = lane & 15;
    const int wave = tid >> 5;
    const int nMt  = CoutP >> 4;
    const int jobs = nMt * 63;             // 63 N-tiles cover g in [0,1008)
    const float* inb = smem + in_off;

    for (int job = wave; job < jobs; job += 8) {
        const int mt = job % nMt;
        const int nt = job / nMt;
        v8f acc;
#pragma unroll
        for (int r = 0; r < 8; ++r) acc[r] = BI[mt * 16 + r + 8 * half];
        const int gbase = nt * 16 + col;

        for (int kc = 0; kc < nkc; ++kc) {
            const int kk0 = kc * 4 + 2 * half;
            const int kk1 = kk0 + 1;
            v2f a, bm;
            a.x = WB[(mt * 16 + col) * Kp + kk0];
            a.y = WB[(mt * 16 + col) * Kp + kk1];
            int ci0 = kk0 / 5; int k0 = kk0 - ci0 * 5; if (ci0 >= Cin) ci0 = Cin - 1;
            int ci1 = kk1 / 5; int k1 = kk1 - ci1 * 5; if (ci1 >= Cin) ci1 = Cin - 1;
            bm.x = inb[ci0 * STR + PADL + gbase + k0 - 2];
            bm.y = inb[ci1 * STR + PADL + gbase + k1 - 2];
            acc = __builtin_amdgcn_wmma_f32_16x16x4_f32(
                false, a, false, bm, (short)0, acc, false, false);
        }

        if (!last) {
            if (gbase < G_GRID) {          // keep g>=1000 tail zero (halo correctness)
                float* outb = smem + out_off;
#pragma unroll
                for (int r = 0; r < 8; ++r) {
                    float v = acc[r];
                    if (relu) v = fmaxf(v, 0.0f);
                    outb[(mt * 16 + r + 8 * half) * STR + PADL + gbase] = v;
                }
            }
        } else {
            // rows 0 (mu) and 1 (pre-softplus sigma) = acc[0],acc[1] of half==0
            if (half == 0 && gbase < G_GRID) {
                float mu = acc[0];
                float x  = acc[1];
                float sp = fmaxf(x, 0.0f) + log1pf(expf(-fabsf(x)));
                yg_out[(b * 2 + 0) * STR + gbase] = mu;
                yg_out[(b * 2 + 1) * STR + gbase] = sp;
            }
        }
    }
}

// One workgroup (256 threads = 8 waves) per batch element; whole conv stack
// LDS-resident (~215 KB of the WGP's 320 KB), intermediates never touch HBM.
__global__ void conv_kernel(const float* __restrict__ hbuf,
                            const float* __restrict__ W1, const float* __restrict__ b1,
                            const float* __restrict__ W2, const float* __restrict__ b2,
                            const float* __restrict__ W3, const float* __restrict__ b3,
                            const float* __restrict__ W4, const float* __restrict__ b4,
                            float* __restrict__ ygrid) {
    extern __shared__ float smem[];
    const int b   = blockIdx.x;
    const int tid = threadIdx.x;
    // zero activation rows: SAME-conv zero halos + zero g>=1000 tails
    for (int i = tid; i < OFF_WB; i += 256) smem[i] = 0.0f;
    __syncthreads();
    // async-copy h (2 x 1000) straight into the padded H2 rows
    for (int i = tid; i < 2 * G_GRID; i += 256) {
        int c = i / G_GRID, g = i - c * G_GRID;
        async_ld_b32(&smem[OFF_H2 + c * STR + PADL + g], hbuf + (b * 2 + c) * STR + g);
    }
    wait_async0();
    conv_layer(smem, OFF_H2,  2,  OFF_A16, 16, 16, W1, b1, true,  false, nullptr, b);
    conv_layer(smem, OFF_A16, 16, OFF_A32, 32, 32, W2, b2, true,  false, nullptr, b);
    conv_layer(smem, OFF_A32, 32, OFF_A16, 16, 16, W3, b3, true,  false, nullptr, b);
    conv_layer(smem, OFF_A16, 16, OFF_A16, 2,  16, W4, b4, false, true,  ygrid,  b);
}

// ---------------------------------------------------------------------------
// Kernel 3: decode set-conv + Gaussian NLL.  One t per thread; async
// double-buffered staging of (mu,sigma) grid chunks; t-grid value computed
// arithmetically (1 FMA beats an LDS load on a TRANS-bound loop).
// ---------------------------------------------------------------------------
__global__ void decode_kernel(const float* __restrict__ xt,
                              const float* __restrict__ yt,
                              const float* __restrict__ ls_rho_p,
                              const float* __restrict__ ygrid,
                              float* __restrict__ out) {
    __shared__ float mus[2][128], sgs[2][128];
    __shared__ float red[128];
    const int b   = blockIdx.y;
    const int tid = threadIdx.x;
    const int t   = blockIdx.x * 128 + tid;
    const float ls = ls_rho_p[0];
    const float c2 = -LOG2E / (2.0f * ls * ls);
    const float dtg = 4.4f / (float)(G_GRID - 1);
    const float x  = xt[b * NT + t];
    const float y  = yt[b * NT + t];
    const float* mub = ygrid + (b * 2 + 0) * STR;
    const float* sgb = ygrid + (b * 2 + 1) * STR;
    float mu = 0.0f, sg = 0.0f;

    const int nchunks = (G_GRID + 127) / 128;     // 8, last partial (104)
    if (tid < G_GRID) {
        async_ld_b32(&mus[0][tid], mub + tid);
        async_ld_b32(&sgs[0][tid], sgb + tid);
    }
    for (int it = 0; it < nchunks; ++it) {
        const int cur = it & 1;
        const int g0  = it * 128;
        wait_async0();
        __syncthreads();
        const int gn = (it + 1) * 128 + tid;
        if (it + 1 < nchunks && gn < G_GRID) {
            async_ld_b32(&mus[cur ^ 1][tid], mub + gn);
            async_ld_b32(&sgs[cur ^ 1][tid], sgb + gn);
        }
        const int jmax = (G_GRID - g0 < 128) ? (G_GRID - g0) : 128;
        for (int j = 0; j < jmax; ++j) {
            float tg = fmaf((float)(g0 + j), dtg, -2.2f);
            float dx = x - tg;
            float w  = exp2f(dx * dx * c2);
            mu += w * mus[cur][j];
            sg += w * sgs[cur][j];
        }
    }
    out[b * NT + t] = mu;                       // mu (B,Nt,1) flat
    out[BATCH * NT + b * NT + t] = sg;          // sigma (B,Nt,1) flat
    float z = (y - mu) / sg;
    float logp = -0.5f * z * z - logf(sg) - 0.5f * LOG_2PI;
    red[tid] = -logp * (1.0f / (float)NT);
    __syncthreads();
    for (int s = 64; s > 0; s >>= 1) {
        if (tid < s) red[tid] += red[tid + s];
        __syncthreads();
    }
    if (tid == 0) atomicAdd(out + 2 * BATCH * NT, red[0]);
}

__global__ void init_loss_kernel(float* __restrict__ out) {
    out[2 * BATCH * NT] = 0.0f;
}

// ---------------------------------------------------------------------------
extern "C" void kernel_launch(void* const* d_in, const int* in_sizes, int n_in,
                              void* d_out, int out_size, void* d_ws, size_t ws_size,
                              hipStream_t stream) {
    const float* xc  = (const float*)d_in[0];
    const float* yc  = (const float*)d_in[1];
    const float* xt  = (const float*)d_in[2];
    const float* yt  = (const float*)d_in[3];
    const float* lsx = (const float*)d_in[4];
    const float* lsr = (const float*)d_in[5];
    const float* W1  = (const float*)d_in[6];
    const float* b1  = (const float*)d_in[7];
    const float* W2  = (const float*)d_in[8];
    const float* b2  = (const float*)d_in[9];
    const float* W3  = (const float*)d_in[10];
    const float* b3  = (const float*)d_in[11];
    const float* W4  = (const float*)d_in[12];
    const float* b4  = (const float*)d_in[13];
    float* out = (float*)d_out;
    float* ws  = (float*)d_ws;
    float* hbuf  = ws;                          // B*2*STR floats
    float* ygrid = ws + BATCH * 2 * STR;        // B*2*STR floats

    encode_kernel<<<dim3(8, BATCH), 128, 0, stream>>>(xc, yc, lsx, hbuf);

    size_t sh = (size_t)SMEM_FLOATS * sizeof(float);
    conv_kernel<<<dim3(BATCH), 256, sh, stream>>>(hbuf, W1, b1, W2, b2, W3, b3,
                                                  W4, b4, ygrid);

    init_loss_kernel<<<1, 1, 0, stream>>>(out);
    decode_kernel<<<dim3(16, BATCH), 128, 0, stream>>>(xt, yt, lsr, ygrid, out);
}